// CoSDynamicAdjacency_44384192037402
// MI455X (gfx1250) — compile-verified
//
#include <hip/hip_runtime.h>
#include <stdint.h>

#define NUM_AGENTS       128
#define TOTAL_K          8
#define ROWS_PER_WAVE    8
#define WAVES_PER_BLOCK  8

typedef float v4f __attribute__((ext_vector_type(4)));

#define SELF_SENTINEL  (-3.402823466e+38f)      // -FLT_MAX: self slot, never selectable
#define REMOVED        (-__builtin_huge_valf()) // -inf: marks a selected/removed slot

// Async copy of one 512B score row (128 f32) into LDS: 32 lanes x 16B each.
// CDNA5 GLOBAL_LOAD_ASYNC_TO_LDS_B128 (GV mode), tracked by ASYNCcnt.
__device__ __forceinline__ void async_load_row128(const float* gsrc, uint32_t lds_off, int lane) {
  uint64_t ga = (uint64_t)(uintptr_t)gsrc + (uint64_t)(lane << 4);
  uint32_t la = lds_off + (uint32_t)(lane << 4);
  asm volatile("global_load_async_to_lds_b128 %0, %1, off"
               :
               : "v"(la), "v"(ga)
               : "memory");
}

__device__ __forceinline__ void s_wait_asynccnt0() {
  asm volatile("s_wait_asynccnt 0" ::: "memory");
}
__device__ __forceinline__ void s_wait_asynccnt1() {
  asm volatile("s_wait_asynccnt 1" ::: "memory");
}

__device__ __forceinline__ float wave_max32(float v) {
#pragma unroll
  for (int off = 16; off >= 1; off >>= 1)
    v = fmaxf(v, __shfl_xor(v, off, 32));
  return v;
}

__device__ __forceinline__ float wave_sum32(float v) {
#pragma unroll
  for (int off = 16; off >= 1; off >>= 1)
    v += __shfl_xor(v, off, 32);
  return v;
}

__global__ __launch_bounds__(WAVES_PER_BLOCK * 32)
void cos_dyn_adj_kernel(const float* __restrict__ scores,
                        float* __restrict__ out,
                        int total_rows) {
  const int lane    = (int)(threadIdx.x & 31u);
  const int wave    = (int)(threadIdx.x >> 5u);
  const int wave_id = blockIdx.x * WAVES_PER_BLOCK + wave;
  const int row0    = wave_id * ROWS_PER_WAVE;
  if (row0 >= total_rows) return;
  const int nrows = min(ROWS_PER_WAVE, total_rows - row0);

  // 2-deep double buffer per wave: 2 x 128 f32 = 1KB/wave, 8KB/block.
  __shared__ float smem[WAVES_PER_BLOCK * 2 * NUM_AGENTS];
  float* buf = &smem[wave * (2 * NUM_AGENTS)];
  const uint32_t buf_off = (uint32_t)(uintptr_t)buf;  // low 32 bits = LDS offset

  async_load_row128(scores + (size_t)row0 * NUM_AGENTS, buf_off, lane);

  const int col0 = lane << 2;  // this lane owns columns [col0, col0+3]

  for (int r = 0; r < nrows; ++r) {
    const int row = row0 + r;
    const int cur = r & 1;

    if (r + 1 < nrows) {
      async_load_row128(scores + (size_t)(row + 1) * NUM_AGENTS,
                        buf_off + (uint32_t)((cur ^ 1) * NUM_AGENTS * 4), lane);
      s_wait_asynccnt1();   // in-order completion -> current buffer ready
    } else {
      s_wait_asynccnt0();
    }

    const float* rowbuf = buf + cur * NUM_AGENTS;
    v4f s = *(const v4f*)(rowbuf + col0);
    const float sc[4] = {s.x, s.y, s.z, s.w};

    const int selfcol = row & (NUM_AGENTS - 1);
    bool iseye[4];
#pragma unroll
    for (int j = 0; j < 4; ++j) iseye[j] = ((col0 + j) == selfcol);

    // ---- softmax over all 128 first (m, p, z available to the top-k loop) ----
    float m = fmaxf(fmaxf(sc[0], sc[1]), fmaxf(sc[2], sc[3]));
    m = wave_max32(m);
    float p[4];
#pragma unroll
    for (int j = 0; j < 4; ++j) p[j] = __expf(sc[j] - m);
    const float z    = wave_sum32(p[0] + p[1] + p[2] + p[3]);
    const float invz = __builtin_amdgcn_rcpf(z);

    // ---- top-7 on RAW scores (exp monotone => same ranking as probs) ----
    float work[4];
#pragma unroll
    for (int j = 0; j < 4; ++j) work[j] = iseye[j] ? SELF_SENTINEL : sc[j];

    float esum = 0.0f;  // sum of exp(sc-m) over selected; bit-equal to owner's p[j]
#pragma unroll
    for (int t = 0; t < TOTAL_K - 1; ++t) {
      float w = fmaxf(fmaxf(work[0], work[1]), fmaxf(work[2], work[3]));
      float bv = w;
#pragma unroll
      for (int off = 16; off >= 1; off >>= 1)
        bv = fmaxf(bv, __shfl_xor(bv, off, 32));
      esum += __expf(bv - m);  // TRANS op, co-executes with the VALU below
      // lowest lane holding the max wins (matches top_k lower-index tie-break)
      unsigned long long ball = __ballot(w == bv);
      int wl = (int)__builtin_ctzll(ball);
      bool live = (lane == wl);
#pragma unroll
      for (int j = 0; j < 4; ++j) {          // remove first matching element
        bool mj = live && (work[j] == bv);
        work[j] = mj ? REMOVED : work[j];
        live = live && !mj;
      }
    }

    // ---- renormalize survivors: sparse = p * invz / (esum*invz + 1e-8) ----
    const float ssum  = esum * invz;
    const float scale = invz * __builtin_amdgcn_rcpf(ssum + 1e-8f);

    bool sel[4];
#pragma unroll
    for (int j = 0; j < 4; ++j) sel[j] = (work[j] == REMOVED);

    v4f sparse;
    sparse.x = sel[0] ? p[0] * scale : 0.0f;
    sparse.y = sel[1] ? p[1] * scale : 0.0f;
    sparse.z = sel[2] ? p[2] * scale : 0.0f;
    sparse.w = sel[3] ? p[3] * scale : 0.0f;

    v4f selfrow;
    selfrow.x = iseye[0] ? 1.0f : 0.0f;
    selfrow.y = iseye[1] ? 1.0f : 0.0f;
    selfrow.z = iseye[2] ? 1.0f : 0.0f;
    selfrow.w = iseye[3] ? 1.0f : 0.0f;

    // ---- 8 x 512B/wave non-temporal b128 stores (output > 192MB L2) ----
    float* obase = out + (size_t)row * (TOTAL_K * NUM_AGENTS);
    __builtin_nontemporal_store(selfrow, (v4f*)(obase + col0));
#pragma unroll
    for (int k = 1; k < TOTAL_K; ++k)
      __builtin_nontemporal_store(sparse, (v4f*)(obase + k * NUM_AGENTS + col0));
  }
}

extern "C" void kernel_launch(void* const* d_in, const int* in_sizes, int n_in,
                              void* d_out, int out_size, void* d_ws, size_t ws_size,
                              hipStream_t stream) {
  const float* scores = (const float*)d_in[0];
  float* out = (float*)d_out;
  const int total_rows = in_sizes[0] / NUM_AGENTS;  // B*N = 65536 rows of 128
  const int rows_per_block = WAVES_PER_BLOCK * ROWS_PER_WAVE;  // 64
  const int blocks = (total_rows + rows_per_block - 1) / rows_per_block;  // 1024
  cos_dyn_adj_kernel<<<blocks, WAVES_PER_BLOCK * 32, 0, stream>>>(scores, out, total_rows);
}